// FCOSPostProcesser_51342039056388
// MI455X (gfx1250) — compile-verified
//
#include <hip/hip_runtime.h>
#include <stdint.h>

// ---------------- constants ----------------
#define NIMG   16
#define NCLS   80
#define NBINS  4096       // float bits [30:19] -> exponent + 4 mantissa bits
#define ACAP   512        // strictly-above list cap (provably < 256 entries)
#define BCAP   4096       // boundary-bin list cap
#define SORTN  4096       // bitonic sort width (power of two)
#define KSEL   256        // NUM_REFINE
#define DEPTH  4          // async prefetch pipeline depth per wave
#define ITERS  16         // 128-element tiles per wave
#define PER_WAVE  (ITERS * 128)
#define PER_BLOCK (8 * PER_WAVE)   // 8 waves (256 threads, wave32)

// level geometry (compile-time)
constexpr int HW_TAB[5]   = {100 * 152, 50 * 76, 25 * 38, 13 * 19, 7 * 10};
constexpr int BASE_TAB[5] = {0, 1216000, 1520000, 1596000, 1615760};

// ws layout (bytes)
#define HIST_BYTES (NIMG * NBINS * 4)            // 262144
#define CTRL_OFF   HIST_BYTES                    // 262144
#define ABOVE_OFF  (CTRL_OFF + 256)              // 262400 (8B aligned)
#define BND_OFF    (ABOVE_OFF + NIMG * ACAP * 8) // 327936

__device__ __forceinline__ int      imin_(int a, int b) { return a < b ? a : b; }
__device__ __forceinline__ unsigned umin_(unsigned a, unsigned b) { return a < b ? a : b; }

// ---- CDNA5 async global->LDS prefetch (ASYNCcnt-tracked) ----
__device__ __forceinline__ void asyncLoad128(unsigned ldsOff, unsigned long long gaddr) {
  asm volatile("global_load_async_to_lds_b128 %0, %1, off"
               :: "v"(ldsOff), "v"(gaddr) : "memory");
}
__device__ __forceinline__ void waitAsync(int n) {
  switch (n) {
    case 0:  asm volatile("s_wait_asynccnt 0" ::: "memory"); break;
    case 1:  asm volatile("s_wait_asynccnt 1" ::: "memory"); break;
    case 2:  asm volatile("s_wait_asynccnt 2" ::: "memory"); break;
    default: asm volatile("s_wait_asynccnt 3" ::: "memory"); break;
  }
}

__device__ __forceinline__ float sigmoidf_(float x) {
  return 1.0f / (1.0f + __expf(-x));
}

// ---------------- pass 1 / pass 3: streaming score kernel ----------------
// MODE 0: build per-image 4096-bin histogram of score bit-patterns.
// MODE 1: compact candidates with bin > B (above) or bin == B (boundary).
// LVL templated so HW / nElems / levelBase are constexpr (div/mod -> mulhi+shift).
template<int MODE, int LVL>
__global__ __launch_bounds__(256) void stream_kernel(
    const float* __restrict__ logits, const float* __restrict__ ctr,
    unsigned* __restrict__ hist, unsigned* __restrict__ ctrl,
    unsigned long long* __restrict__ above, unsigned long long* __restrict__ bnd)
{
  constexpr int HW        = HW_TAB[LVL];
  constexpr int nElems    = HW * NCLS;
  constexpr int levelBase = BASE_TAB[LVL];

  __shared__ __align__(16) float stage[8][DEPTH][128];   // 16 KB staging
  __shared__ unsigned shHist[NBINS];                     // 16 KB histogram
  const int tid  = threadIdx.x;
  const int w    = tid >> 5;
  const int lane = tid & 31;
  const int n    = blockIdx.y;

  if (MODE == 0) {
    for (int b = tid; b < NBINS; b += 256) shHist[b] = 0u;
    __syncthreads();
  }
  unsigned B = 0;
  if (MODE == 1) B = ctrl[n * 4 + 2];

  const float* lg = logits + (long long)n * nElems;   // [C*HW] for this image
  const float* ct = ctr    + (long long)n * HW;       // [HW]

  auto process = [&](int e, float x) {
    // native layout index e = c*HW + hw ; reference flat idx = hw*C + c
    float s = sigmoidf_(x);
    float v = 0.0f;
    if (s > 0.05f) {
      unsigned hw = (unsigned)e % (unsigned)HW;       // constant divisor
      v = s * sigmoidf_(ct[hw]);
    }
    unsigned bits = __float_as_uint(v);   // v >= 0 -> order-preserving key
    unsigned bin  = bits >> 19;           // [0, 4095]
    if (MODE == 0) {
      atomicAdd(&shHist[bin], 1u);
    } else {
      if (bin > B || (bin == B && bits != 0u)) {
        unsigned c   = (unsigned)e / (unsigned)HW;    // constant divisor
        unsigned hw  = (unsigned)e - c * (unsigned)HW;
        unsigned idx = (unsigned)levelBase + hw * NCLS + c;
        unsigned long long key =
            ((unsigned long long)bits << 32) | (unsigned)(~idx); // score desc, idx asc
        if (bin > B) {
          unsigned p = atomicAdd(&ctrl[n * 4 + 0], 1u);
          if (p < ACAP) above[(size_t)n * ACAP + p] = key;
        } else {
          unsigned p = atomicAdd(&ctrl[n * 4 + 1], 1u);
          if (p < BCAP) bnd[(size_t)n * BCAP + p] = key;
        }
      }
    }
  };

  // per-wave contiguous chunk; full 128-elem tiles go through the async pipe
  const int waveBase = (blockIdx.x * 8 + w) * PER_WAVE;
  int nf = 0;
  if (waveBase < nElems) nf = imin_(ITERS, (nElems - waveBase) >> 7);

  const int pre = imin_(nf, DEPTH);
  for (int p = 0; p < pre; ++p) {
    unsigned ldsOff = (unsigned)(uintptr_t)&stage[w][p][lane * 4];
    asyncLoad128(ldsOff,
        (unsigned long long)(uintptr_t)(lg + waveBase + p * 128 + lane * 4));
  }
  for (int i = 0; i < nf; ++i) {
    waitAsync(imin_(nf - i - 1, DEPTH - 1));  // oldest in-flight tile retired
    const int slot = i & (DEPTH - 1);
    float4 v4 = *(const float4*)&stage[w][slot][lane * 4];
    const int nx = i + DEPTH;
    if (nx < nf) {
      unsigned ldsOff = (unsigned)(uintptr_t)&stage[w][slot][lane * 4];
      asyncLoad128(ldsOff,
          (unsigned long long)(uintptr_t)(lg + waveBase + nx * 128 + lane * 4));
    }
    const int eb = waveBase + i * 128 + lane * 4;
    process(eb + 0, v4.x); process(eb + 1, v4.y);
    process(eb + 2, v4.z); process(eb + 3, v4.w);
  }
  // scalar tail (partial tile / end of tensor)
  const int tailBase = waveBase + nf * 128;
  const int waveEnd  = imin_(waveBase + PER_WAVE, nElems);
  for (int e = tailBase + lane; e < waveEnd; e += 32) process(e, lg[e]);

  if (MODE == 0) {
    __syncthreads();
    for (int b = tid; b < NBINS; b += 256) {
      unsigned v = shHist[b];
      if (v) atomicAdd(&hist[(size_t)n * NBINS + b], v);
    }
  }
}

// ---------------- pass 2: find the boundary bin per image ----------------
__global__ __launch_bounds__(256) void scan_kernel(const unsigned* __restrict__ hist,
                                                   unsigned* __restrict__ ctrl)
{
  const int n = blockIdx.x;
  __shared__ unsigned h[NBINS];
  for (int b = threadIdx.x; b < NBINS; b += blockDim.x) h[b] = hist[(size_t)n * NBINS + b];
  __syncthreads();
  if (threadIdx.x == 0) {
    unsigned sum = 0, B = 0;
    for (int b = NBINS - 1; b >= 0; --b) {
      if (sum + h[b] >= (unsigned)KSEL) { B = (unsigned)b; break; }
      sum += h[b];
    }
    ctrl[n * 4 + 2] = B;    // boundary bin
    ctrl[n * 4 + 3] = sum;  // count strictly above (debug)
  }
}

// ---------------- pass 4: sort survivors, decode boxes, write output ----------------
struct LevelP { const float* reg; const float* loc; int HW; int base; float stride; };
struct P4 { LevelP l[5]; };

__global__ __launch_bounds__(1024) void final_kernel(
    P4 p, const unsigned long long* __restrict__ above,
    const unsigned long long* __restrict__ bnd,
    const unsigned* __restrict__ ctrl, float* __restrict__ out)
{
  const int n = blockIdx.x, tid = threadIdx.x;
  __shared__ unsigned long long keys[SORTN];   // 32 KB

  unsigned nA = umin_(ctrl[n * 4 + 0], (unsigned)ACAP);   // < 256 by construction
  unsigned nB = umin_(ctrl[n * 4 + 1], (unsigned)BCAP);
  if (nA + nB > (unsigned)SORTN) nB = (unsigned)SORTN - nA;

  for (int i = tid; i < SORTN; i += 1024) {
    unsigned long long kk = 0ull;
    if (i < (int)nA)              kk = above[(size_t)n * ACAP + i];
    else if (i < (int)(nA + nB))  kk = bnd[(size_t)n * BCAP + (i - nA)];
    keys[i] = kk;
  }

  // bitonic sort, descending (key = score bits : ~idx)
  for (unsigned kk = 2; kk <= (unsigned)SORTN; kk <<= 1) {
    for (unsigned j = kk >> 1; j > 0; j >>= 1) {
      __syncthreads();
      for (unsigned i = tid; i < (unsigned)SORTN; i += 1024) {
        unsigned ixj = i ^ j;
        if (ixj > i) {
          unsigned long long a = keys[i], b = keys[ixj];
          bool desc = ((i & kk) == 0);
          if (desc ? (a < b) : (a > b)) { keys[i] = b; keys[ixj] = a; }
        }
      }
    }
  }
  __syncthreads();

  if (tid < KSEL) {
    unsigned long long key = keys[tid];
    unsigned bits = (unsigned)(key >> 32);
    float sc = 0.0f, b0 = 0.0f, b1 = 0.0f, b2 = 0.0f, b3 = 0.0f;
    int cls = 0, lvl = 0;
    if (bits != 0u) {
      float v = __uint_as_float(bits);
      sc = sqrtf(v);
      unsigned idx = ~(unsigned)(key & 0xFFFFFFFFull);
      if      (idx >= (unsigned)p.l[4].base) lvl = 4;
      else if (idx >= (unsigned)p.l[3].base) lvl = 3;
      else if (idx >= (unsigned)p.l[2].base) lvl = 2;
      else if (idx >= (unsigned)p.l[1].base) lvl = 1;
      else                                   lvl = 0;
      const LevelP L = p.l[lvl];
      unsigned rel = idx - (unsigned)L.base;
      unsigned hw  = rel / (unsigned)NCLS;   // constant divisor (80)
      cls = (int)(rel - hw * (unsigned)NCLS);
      float px = L.loc[hw * 2 + 0], py = L.loc[hw * 2 + 1];
      const float* rg = L.reg + (size_t)n * 4 * L.HW + hw;  // [N,4,H,W]
      float r0 = rg[0]        * L.stride;
      float r1 = rg[L.HW]     * L.stride;
      float r2 = rg[2 * L.HW] * L.stride;
      float r3 = rg[3 * L.HW] * L.stride;
      b0 = px - r0; b1 = py - r1; b2 = px + r2; b3 = py + r3;
    }
    const int o = n * KSEL + tid;
    out[o] = sc;                                  // top_sc  [16,256]
    float* ob = out + NIMG * KSEL;                // top_bx  [16,256,4]
    ob[o * 4 + 0] = b0; ob[o * 4 + 1] = b1;
    ob[o * 4 + 2] = b2; ob[o * 4 + 3] = b3;
    int* oc = (int*)(out + NIMG * KSEL * 5);      // top_cl  [16,256] int32
    oc[o] = cls;
    int* ol = (int*)(out + NIMG * KSEL * 6);      // top_lv  [16,256] int32
    ol[o] = lvl;
  }
}

// ---------------- host launcher ----------------
template<int MODE>
static void launch_level(int l, const float* lg, const float* ct,
                         unsigned* hist, unsigned* ctrl,
                         unsigned long long* above, unsigned long long* bnd,
                         hipStream_t stream) {
  const int nElems = HW_TAB[l] * NCLS;
  const int blocks = (nElems + PER_BLOCK - 1) / PER_BLOCK;
  dim3 grid(blocks, NIMG);
  switch (l) {
    case 0: stream_kernel<MODE, 0><<<grid, 256, 0, stream>>>(lg, ct, hist, ctrl, above, bnd); break;
    case 1: stream_kernel<MODE, 1><<<grid, 256, 0, stream>>>(lg, ct, hist, ctrl, above, bnd); break;
    case 2: stream_kernel<MODE, 2><<<grid, 256, 0, stream>>>(lg, ct, hist, ctrl, above, bnd); break;
    case 3: stream_kernel<MODE, 3><<<grid, 256, 0, stream>>>(lg, ct, hist, ctrl, above, bnd); break;
    default: stream_kernel<MODE, 4><<<grid, 256, 0, stream>>>(lg, ct, hist, ctrl, above, bnd); break;
  }
}

extern "C" void kernel_launch(void* const* d_in, const int* in_sizes, int n_in,
                              void* d_out, int out_size, void* d_ws, size_t ws_size,
                              hipStream_t stream) {
  (void)in_sizes; (void)n_in; (void)out_size; (void)ws_size;

  static const float STR[5] = {8.f, 16.f, 32.f, 64.f, 128.f};

  char* ws = (char*)d_ws;
  unsigned*           hist  = (unsigned*)ws;
  unsigned*           ctrl  = (unsigned*)(ws + CTRL_OFF);
  unsigned long long* above = (unsigned long long*)(ws + ABOVE_OFF);
  unsigned long long* bnd   = (unsigned long long*)(ws + BND_OFF);

  // zero histograms + counters each call (graph-capture safe)
  hipMemsetAsync(ws, 0, ABOVE_OFF, stream);

  // pass 1: histogram of score keys
  for (int l = 0; l < 5; ++l)
    launch_level<0>(l, (const float*)d_in[4 * l + 0], (const float*)d_in[4 * l + 2],
                    hist, ctrl, above, bnd, stream);
  // pass 2: find boundary bin
  scan_kernel<<<NIMG, 256, 0, stream>>>(hist, ctrl);
  // pass 3: compact survivors
  for (int l = 0; l < 5; ++l)
    launch_level<1>(l, (const float*)d_in[4 * l + 0], (const float*)d_in[4 * l + 2],
                    hist, ctrl, above, bnd, stream);
  // pass 4: sort top candidates, decode, emit
  P4 p;
  for (int l = 0; l < 5; ++l) {
    p.l[l].reg = (const float*)d_in[4 * l + 1];
    p.l[l].loc = (const float*)d_in[4 * l + 3];
    p.l[l].HW = HW_TAB[l]; p.l[l].base = BASE_TAB[l]; p.l[l].stride = STR[l];
  }
  final_kernel<<<NIMG, 1024, 0, stream>>>(p, above, bnd, ctrl, (float*)d_out);
}